// PositionWiseAttention_71640054497426
// MI455X (gfx1250) — compile-verified
//
#include <hip/hip_runtime.h>

// CDNA5 / gfx1250: wave32, fp32 WMMA 16x16x4, TDM async tensor loads.
typedef __attribute__((ext_vector_type(2))) float v2f;
typedef __attribute__((ext_vector_type(8))) float v8f;
typedef __attribute__((ext_vector_type(4))) unsigned int u32x4;
typedef __attribute__((ext_vector_type(4))) int i32x4;
typedef __attribute__((ext_vector_type(8))) int i32x8;

#define WB   4
#define WC   256
#define WH   64
#define WW   64
#define WN   4096     // H*W
#define WMID 64
#define NSTAT 512     // softmax-stat blocks per batch (512 = 2*256 -> fixed k-tile per block)

#if __has_include(<hip/amd_detail/amd_gfx1250_TDM.h>)
#define TDM_6ARG 1
#endif

__device__ __forceinline__ v8f wmma_f32(v2f a, v2f b, v8f c) {
  return __builtin_amdgcn_wmma_f32_16x16x4_f32(false, a, false, b, (short)0, c,
                                               false, false);
}

// Tensor Data Mover: async 2D tile (tile_w x tile_h f32, row stride
// stride_elems) from global -> LDS. Optional pad: +1 dword per 16 dwords
// (stride-17 rows in LDS, bank-conflict-free column reads).
__device__ __forceinline__ void tdm_load_2d(unsigned lds_off, const float* gptr,
                                            unsigned tile_w, unsigned tile_h,
                                            unsigned stride_elems, bool pad16) {
  const unsigned long long ga = (unsigned long long)(uintptr_t)gptr;
  u32x4 g0;
  g0[0] = 1u;                                            // count=1, no gather
  g0[1] = lds_off;                                       // LDS byte address
  g0[2] = (unsigned)ga;                                  // global_addr[31:0]
  g0[3] = (unsigned)((ga >> 32) & 0x01ffffffULL) | (2u << 30);  // addr[56:32] | type=2
  i32x8 g1;
  unsigned dw0 = (2u << 16);                             // data_size = 4B
  if (pad16) dw0 |= (1u << 20) | (3u << 22);             // pad_en, 16 dwords, +1 dword
  g1[0] = (int)dw0;
  g1[1] = (int)(((unsigned)WN & 0xffffu) << 16);         // tensor_dim0[15:0]
  g1[2] = (int)((((unsigned)WN >> 16) & 0xffffu) | ((tile_h & 0xffffu) << 16));
  g1[3] = (int)((tile_w & 0xffffu) << 16);               // tensor_dim1 hi | tile_dim0
  g1[4] = (int)(tile_h & 0xffffu);                       // tile_dim1 | tile_dim2=0
  g1[5] = (int)stride_elems;                             // tensor_dim0_stride[31:0]
  g1[6] = 0;
  g1[7] = 0;
  const i32x4 z4 = {0, 0, 0, 0};
#ifdef TDM_6ARG
  const i32x8 z8 = {0, 0, 0, 0, 0, 0, 0, 0};
  __builtin_amdgcn_tensor_load_to_lds(g0, g1, z4, z4, z8, 0);
#else
  __builtin_amdgcn_tensor_load_to_lds(g0, g1, z4, z4, 0);
#endif
}

#define LDS_OFF(p) ((unsigned)(uintptr_t)(p))

// ---------------------------------------------------------------------------
// K1: a = Wa @ x, b = Wb @ x  (1x1 conv == GEMM 64x256 * 256x4096 per batch)
// ---------------------------------------------------------------------------
__global__ void k_proj_ab(const float* __restrict__ x,
                          const float* __restrict__ wa, const float* __restrict__ ba,
                          const float* __restrict__ wb, const float* __restrict__ bb,
                          float* __restrict__ outa, float* __restrict__ outb) {
  const int lane = threadIdx.x & 31;
  const int col  = lane & 15;
  const int half = lane >> 4;
  const int n0 = blockIdx.x * 16;
  const int m0 = blockIdx.y * 16;
  const int b  = blockIdx.z >> 1;
  const bool second = blockIdx.z & 1;
  const float* w    = second ? wb : wa;
  const float* bias = second ? bb : ba;
  float* out        = second ? outb : outa;
  const float* xb = x + (size_t)b * WC * WN;

  v8f acc = {};
  for (int k0 = 0; k0 < WC; k0 += 4) {
    const int kk = k0 + 2 * half;
    v2f A, Bv;
    A.x  = w[(size_t)(m0 + col) * WC + kk];
    A.y  = w[(size_t)(m0 + col) * WC + kk + 1];
    Bv.x = xb[(size_t)kk * WN + n0 + col];
    Bv.y = xb[(size_t)(kk + 1) * WN + n0 + col];
    __builtin_prefetch(xb + (size_t)(kk + 4) * WN + n0 + col, 0, 1);
    acc = wmma_f32(A, Bv, acc);
  }
  float* ob = out + (size_t)b * WMID * WN;
#pragma unroll
  for (int j = 0; j < 8; ++j) {
    const int m = m0 + j + 8 * half;
    ob[(size_t)m * WN + n0 + col] = acc[j] + bias[m];
  }
}

// ---------------------------------------------------------------------------
// K2/K5: 3x3 conv (pad 1) as 9 shifted GEMMs, 16 outch x 16 pixel tile/wave.
// ---------------------------------------------------------------------------
__global__ void k_conv3x3(const float* __restrict__ xin, const float* __restrict__ xres,
                          const float* __restrict__ w, const float* __restrict__ bias,
                          float* __restrict__ out) {
  const int lane = threadIdx.x & 31;
  const int col  = lane & 15;
  const int half = lane >> 4;
  const int n0  = blockIdx.x * 16;
  const int oc0 = blockIdx.y * 16;
  const int b   = blockIdx.z;
  const int h   = n0 >> 6;
  const int w0  = n0 & 63;
  const float* xb = xin + (size_t)b * WC * WN;
  const float* xr = xres ? xres + (size_t)b * WC * WN : nullptr;

  v8f acc = {};
  for (int kh = 0; kh < 3; ++kh) {
    const int hs = h + kh - 1;
    const bool hok = (unsigned)hs < (unsigned)WH;
    for (int kw = 0; kw < 3; ++kw) {
      const int wsx = w0 + col + kw - 1;
      const bool ok = hok && ((unsigned)wsx < (unsigned)WW);
      const size_t pix = (size_t)hs * WW + wsx;
      for (int k0 = 0; k0 < WC; k0 += 4) {
        const int ic = k0 + 2 * half;
        v2f A, Bv;
        const float* wp = w + ((size_t)(oc0 + col) * WC + ic) * 9 + kh * 3 + kw;
        A.x = wp[0];
        A.y = wp[9];
        if (ok) {
          const float* xp = xb + (size_t)ic * WN + pix;
          Bv.x = xp[0];
          Bv.y = xp[WN];
          if (xr) {
            const float* rp = xr + (size_t)ic * WN + pix;
            Bv.x += rp[0];
            Bv.y += rp[WN];
          }
        } else {
          Bv.x = 0.f;
          Bv.y = 0.f;
        }
        acc = wmma_f32(A, Bv, acc);
      }
    }
  }
  float* ob = out + (size_t)b * WC * WN;
#pragma unroll
  for (int j = 0; j < 8; ++j) {
    const int oc = oc0 + j + 8 * half;
    ob[(size_t)oc * WN + n0 + col] = acc[j] + bias[oc];
  }
}

// ---------------------------------------------------------------------------
// K3: fused softmax pass 1. Per block: fixed k-tile (b-tile TDM-staged once),
// a-tile double-buffered via async TDM prefetch overlapping the WMMA build.
// ---------------------------------------------------------------------------
__global__ void k_softmax_stats(const float* __restrict__ amat, const float* __restrict__ bmat,
                                float* __restrict__ pmax, float* __restrict__ psum) {
  __shared__ float sb[WMID * 16];        // fixed b-tile  (4 KB)
  __shared__ float sa[2][WMID * 16];     // a-tile double buffer (8 KB)
  const int lane = threadIdx.x & 31;
  const int col  = lane & 15;
  const int half = lane >> 4;
  const int s = blockIdx.x;              // 0..NSTAT-1
  const int b = blockIdx.y;
  const float* ab = amat + (size_t)b * WMID * WN;
  const float* bb = bmat + (size_t)b * WMID * WN;
  const int k0  = (s & 255) * 16;        // fixed k-tile for this block
  const int nt0 = (s >> 8);              // n-tile sequence: nt0 + 2*i

  tdm_load_2d(LDS_OFF(sb), bb + k0, 16, WMID, WN, false);
  tdm_load_2d(LDS_OFF(sa[0]), ab + nt0 * 16, 16, WMID, WN, false);

  float mx = -__builtin_inff();
  float sm = 0.f;
  const int iters = (256 * 256) / NSTAT;          // 128
  for (int i = 0; i < iters; ++i) {
    const int buf = i & 1;
    if (i + 1 < iters) {
      tdm_load_2d(LDS_OFF(sa[buf ^ 1]), ab + (nt0 + 2 * (i + 1)) * 16, 16, WMID, WN, false);
      __builtin_amdgcn_s_wait_tensorcnt(1);       // current tiles done, prefetch in flight
    } else {
      __builtin_amdgcn_s_wait_tensorcnt(0);
    }
    const float* at = sa[buf];
    v8f p = {};
    for (int m0 = 0; m0 < WMID; m0 += 4) {
      const int mm = m0 + 2 * half;
      v2f A, Bv;
      A.x  = at[mm * 16 + col];                   // A[n][m] = a[m][n]
      A.y  = at[(mm + 1) * 16 + col];
      Bv.x = sb[mm * 16 + col];                   // B[m][k] = b[m][k]
      Bv.y = sb[(mm + 1) * 16 + col];
      p = wmma_f32(A, Bv, p);
    }
    float lm = p[0];
#pragma unroll
    for (int j = 1; j < 8; ++j) lm = fmaxf(lm, p[j]);
    for (int off = 16; off; off >>= 1) lm = fmaxf(lm, __shfl_xor(lm, off, 32));
    float ls = 0.f;
#pragma unroll
    for (int j = 0; j < 8; ++j) ls += __expf(p[j] - lm);
    for (int off = 16; off; off >>= 1) ls += __shfl_xor(ls, off, 32);
    const float nm = fmaxf(mx, lm);
    sm = sm * __expf(mx - nm) + ls * __expf(lm - nm);
    mx = nm;
  }
  if (lane == 0) {
    pmax[b * NSTAT + s] = mx;
    psum[b * NSTAT + s] = sm;
  }
}

// K3b: combine per-block partials into per-batch {gmax, 1/gsum}.
__global__ void k_stats_reduce(const float* __restrict__ pmax, const float* __restrict__ psum,
                               float* __restrict__ stats) {
  __shared__ float smax[256];
  __shared__ float ssum[256];
  const int b = blockIdx.x;
  const int tid = threadIdx.x;
  float m = -__builtin_inff();
  for (int i = tid; i < NSTAT; i += 256) m = fmaxf(m, pmax[b * NSTAT + i]);
  smax[tid] = m;
  __syncthreads();
  for (int st = 128; st; st >>= 1) {
    if (tid < st) smax[tid] = fmaxf(smax[tid], smax[tid + st]);
    __syncthreads();
  }
  const float gmax = smax[0];
  __syncthreads();
  float sacc = 0.f;
  for (int i = tid; i < NSTAT; i += 256)
    sacc += psum[b * NSTAT + i] * __expf(pmax[b * NSTAT + i] - gmax);
  ssum[tid] = sacc;
  __syncthreads();
  for (int st = 128; st; st >>= 1) {
    if (tid < st) ssum[tid] += ssum[tid + st];
    __syncthreads();
  }
  if (tid == 0) {
    stats[b * 2]     = gmax;
    stats[b * 2 + 1] = 1.f / ssum[0];
  }
}

// ---------------------------------------------------------------------------
// K4: pass 2 — out[b,n,ch] = sum_k softmax(p)[n,k] * c[ch,k].
// Wave 0: a-tile staged once; b-tile + 64ch x 16k c-tile double-buffered via
// TDM (c-tile uses TDM pad -> stride-17 rows, conflict-free column reads).
// Wave 0 rebuilds+softmaxes the p tile into LDS; all 4 waves consume it.
// ---------------------------------------------------------------------------
__global__ void k_attn_out(const float* __restrict__ amat, const float* __restrict__ bmat,
                           const float* __restrict__ cmat, const float* __restrict__ stats,
                           float* __restrict__ outmid) {
  __shared__ float atile[WMID * 16];       // 4 KB
  __shared__ float btile[2][WMID * 16];    // 8 KB
  __shared__ float ctile[2][64 * 17];      // 8.5 KB (TDM-padded rows)
  __shared__ float ptile[16 * 17];         // 1 KB
  const int lane = threadIdx.x & 31;
  const int wave = threadIdx.x >> 5;       // 0..3
  const int col  = lane & 15;
  const int half = lane >> 4;
  const int n0  = blockIdx.x * 16;
  const int chb = blockIdx.y * 64;         // 64-channel slab per block
  const int ch0 = chb + wave * 16;
  const int b   = blockIdx.z;
  const float* ab = amat + (size_t)b * WMID * WN;
  const float* bb = bmat + (size_t)b * WMID * WN;
  const float* cb = cmat + (size_t)b * WC * WN + (size_t)chb * WN;
  const float gmax = stats[b * 2];
  const float inv  = stats[b * 2 + 1];

  if (wave == 0) {
    tdm_load_2d(LDS_OFF(atile), ab + n0, 16, WMID, WN, false);
    tdm_load_2d(LDS_OFF(btile[0]), bb, 16, WMID, WN, false);
    tdm_load_2d(LDS_OFF(ctile[0]), cb, 16, 64, WN, true);
  }

  v8f acc = {};
  for (int k0 = 0; k0 < WN; k0 += 16) {
    const int buf = (k0 >> 4) & 1;
    if (wave == 0) {
      if (k0 + 16 < WN) {
        tdm_load_2d(LDS_OFF(btile[buf ^ 1]), bb + k0 + 16, 16, WMID, WN, false);
        tdm_load_2d(LDS_OFF(ctile[buf ^ 1]), cb + k0 + 16, 16, 64, WN, true);
        __builtin_amdgcn_s_wait_tensorcnt(2);   // current a/b/c done, 2 prefetches fly
      } else {
        __builtin_amdgcn_s_wait_tensorcnt(0);
      }
      const float* bt = btile[buf];
      v8f p = {};
      for (int m0 = 0; m0 < WMID; m0 += 4) {
        const int mm = m0 + 2 * half;
        v2f A, Bv;
        A.x  = atile[mm * 16 + col];
        A.y  = atile[(mm + 1) * 16 + col];
        Bv.x = bt[mm * 16 + col];
        Bv.y = bt[(mm + 1) * 16 + col];
        p = wmma_f32(A, Bv, p);
      }
#pragma unroll
      for (int j = 0; j < 8; ++j) {
        const int n = j + 8 * half;
        ptile[n * 17 + col] = __expf(p[j] - gmax) * inv;
      }
    }
    __syncthreads();                            // ptile + ctile[buf] visible to all
    const float* ct = ctile[buf];
#pragma unroll
    for (int s2 = 0; s2 < 4; ++s2) {
      const int kk = s2 * 4 + 2 * half;
      v2f A, Bv;
      A.x  = ptile[col * 17 + kk];              // A[n][k] (LDS transpose)
      A.y  = ptile[col * 17 + kk + 1];
      Bv.x = ct[(wave * 16 + col) * 17 + kk];   // B[k][ch] = c[ch][k], padded rows
      Bv.y = ct[(wave * 16 + col) * 17 + kk + 1];
      acc = wmma_f32(A, Bv, acc);
    }
    __syncthreads();                            // before buffers are overwritten
  }
  float* ob = outmid + (size_t)b * WN * WC;
#pragma unroll
  for (int j = 0; j < 8; ++j) {
    const int n = n0 + j + 8 * half;
    ob[(size_t)n * WC + ch0 + col] = acc[j];
  }
}

// ---------------------------------------------------------------------------
extern "C" void kernel_launch(void* const* d_in, const int* in_sizes, int n_in,
                              void* d_out, int out_size, void* d_ws, size_t ws_size,
                              hipStream_t stream) {
  const float* x     = (const float*)d_in[0];
  const float* paw   = (const float*)d_in[1];
  const float* pab   = (const float*)d_in[2];
  const float* pbw   = (const float*)d_in[3];
  const float* pbb   = (const float*)d_in[4];
  const float* pw    = (const float*)d_in[5];
  const float* pbias = (const float*)d_in[6];
  const float* ow    = (const float*)d_in[7];
  const float* obias = (const float*)d_in[8];
  float* out = (float*)d_out;

  float* wsa   = (float*)d_ws;
  float* wsb   = wsa + (size_t)WB * WMID * WN;
  float* wsc   = wsb + (size_t)WB * WMID * WN;
  float* wsm   = wsc + (size_t)WB * WC * WN;
  float* wpmax = wsm + (size_t)WB * WC * WN;
  float* wpsum = wpmax + WB * NSTAT;
  float* wstat = wpsum + WB * NSTAT;

  k_proj_ab<<<dim3(256, 4, WB * 2), 32, 0, stream>>>(x, paw, pab, pbw, pbb, wsa, wsb);
  k_conv3x3<<<dim3(256, 16, WB), 32, 0, stream>>>(x, nullptr, pw, pbias, wsc);
  k_softmax_stats<<<dim3(NSTAT, WB), 32, 0, stream>>>(wsa, wsb, wpmax, wpsum);
  k_stats_reduce<<<dim3(WB), 256, 0, stream>>>(wpmax, wpsum, wstat);
  k_attn_out<<<dim3(256, 4, WB), 128, 0, stream>>>(wsa, wsb, wsc, wstat, wsm);
  k_conv3x3<<<dim3(256, 16, WB), 32, 0, stream>>>(wsm, x, ow, obias, out);
}